// LocScaleTransform_49555332661372
// MI455X (gfx1250) — compile-verified
//
#include <hip/hip_runtime.h>
#include <stdint.h>

// Problem constants (match reference): B, H, W, C
#define BB 16
#define HH 256
#define WW 256
#define CC 64
#define ROW_ELTS (WW * CC)              // 16384 floats per source row
#define ROW_BYTES (ROW_ELTS * 4)        // 65536 bytes

typedef uint32_t u32x4 __attribute__((ext_vector_type(4)));
typedef int      i32x4 __attribute__((ext_vector_type(4)));
typedef int      i32x8 __attribute__((ext_vector_type(8)));

// Issue one TDM DMA: copy a full 64 KB image row (W*C f32, contiguous) from
// global memory into LDS at byte offset lds_byte_off. 2D descriptor:
//   data_size=4B, tensor_dim0=tile_dim0=16384, tile_dim1=1.
__device__ __forceinline__ void tdm_load_row(const float* __restrict__ src,
                                             uint32_t lds_byte_off) {
    uint64_t ga = (uint64_t)(uintptr_t)src;
    u32x4 g0;
    g0[0] = 1u;                                       // count=1, user mode, no gather
    g0[1] = lds_byte_off;                             // D#.lds_addr (bytes)
    g0[2] = (uint32_t)(ga & 0xFFFFFFFFu);             // global_addr[31:0]
    g0[3] = (uint32_t)((ga >> 32) & 0x01FFFFFFu)      // global_addr[56:32]
          | (2u << 30);                               // type = 2 ("image")
    i32x8 g1;
    g1[0] = 0x00020000;                               // data_size = 4B; no multicast
    g1[1] = (int)(((uint32_t)ROW_ELTS & 0xFFFFu) << 16); // tensor_dim0 lo16 @ [63:48]
    g1[2] = 0x00010000;                               // tensor_dim0 hi16=0; tensor_dim1=1
    g1[3] = (int)(((uint32_t)ROW_ELTS & 0xFFFFu) << 16); // tile_dim0 @ [127:112]
    g1[4] = 1;                                        // tile_dim1 = 1
    g1[5] = ROW_ELTS;                                 // tensor_dim0_stride
    g1[6] = 0;
    g1[7] = 0;
    i32x4 z4 = {0, 0, 0, 0};
#if defined(__clang_major__) && (__clang_major__ >= 23)
    i32x8 z8 = {0, 0, 0, 0, 0, 0, 0, 0};
    __builtin_amdgcn_tensor_load_to_lds(g0, g1, z4, z4, z8, 0);
#else
    __builtin_amdgcn_tensor_load_to_lds(g0, g1, z4, z4, 0);
#endif
}

__global__ __launch_bounds__(256) void
LocScaleTransform_bilinear_kernel(const float* __restrict__ transforms,
                                  const float* __restrict__ imgs,
                                  float* __restrict__ out) {
    extern __shared__ __align__(16) float smem[];     // 2 * 16384 floats = 128 KB
    const int row = blockIdx.x;                       // output row, 0..255
    const int b   = blockIdx.y;                       // batch, 0..15
    const int tid = threadIdx.x;

    const float tx = transforms[b * 3 + 0];
    const float ty = transforms[b * 3 + 1];
    const float sc = transforms[b * 3 + 2] + 1.0f;

    // y sample for this output row: linspace(-1,1,256)[row]*scale + ty, then map to texels
    const float gy  = 2.0f * (float)row * (1.0f / 255.0f) - 1.0f;
    const float fy  = (gy * sc + ty + 1.0f) * 128.0f;
    float y0f = floorf(fy);
    float y1f = y0f + 1.0f;
    y0f = fminf(fmaxf(y0f, 0.0f), (float)(HH - 1));
    y1f = fminf(fmaxf(y1f, 0.0f), (float)(HH - 1));
    const float wyT = y1f - fy;                       // weight of row y0 (clipped, per ref)
    const float wyB = fy - y0f;                       // weight of row y1
    const int y0 = (int)y0f;
    const int y1 = (int)y1f;

    // Stage source rows y0 and y1 into LDS via the Tensor Data Mover (wave 0 only;
    // TDM ignores EXEC, so gate by a wave-uniform branch to issue each DMA once).
    if (tid < 32) {
        const float* base = imgs + (size_t)b * (size_t)HH * ROW_ELTS;
        tdm_load_row(base + (size_t)y0 * ROW_ELTS, 0u);
        tdm_load_row(base + (size_t)y1 * ROW_ELTS, (uint32_t)ROW_BYTES);
        __builtin_amdgcn_s_wait_tensorcnt(0);
    }
    __syncthreads();

    const float4* __restrict__ r0 = (const float4*)smem;               // row y0
    const float4* __restrict__ r1 = (const float4*)(smem + ROW_ELTS);  // row y1
    float4* __restrict__ o4 =
        (float4*)(out + ((size_t)b * HH + row) * (size_t)ROW_ELTS);

    // 256 pixels * 16 float4-chunks = 4096 float4 per row; 16 per thread.
    // 16 consecutive threads cooperate on one pixel (consecutive channel chunks)
    // -> conflict-free ds_load_b128 and fully coalesced global_store_b128.
#pragma unroll 4
    for (int i = 0; i < 16; ++i) {
        const int k   = tid + (i << 8);
        const int col = k >> 4;
        const int c4  = k & 15;

        const float gx  = 2.0f * (float)col * (1.0f / 255.0f) - 1.0f;
        const float fx  = (gx * sc + tx + 1.0f) * 128.0f;
        float x0f = floorf(fx);
        float x1f = x0f + 1.0f;
        x0f = fminf(fmaxf(x0f, 0.0f), (float)(WW - 1));
        x1f = fminf(fmaxf(x1f, 0.0f), (float)(WW - 1));
        const float wxL = x1f - fx;
        const float wxR = fx - x0f;
        const int x0 = (int)x0f;
        const int x1 = (int)x1f;

        const float wa = wyT * wxL;
        const float wb = wyT * wxR;
        const float wc = wyB * wxL;
        const float wd = wyB * wxR;

        const float4 Ia = r0[x0 * 16 + c4];
        const float4 Ib = r0[x1 * 16 + c4];
        const float4 Ic = r1[x0 * 16 + c4];
        const float4 Id = r1[x1 * 16 + c4];

        float4 o;
        o.x = wa * Ia.x + wb * Ib.x + wc * Ic.x + wd * Id.x;
        o.y = wa * Ia.y + wb * Ib.y + wc * Ic.y + wd * Id.y;
        o.z = wa * Ia.z + wb * Ib.z + wc * Ic.z + wd * Id.z;
        o.w = wa * Ia.w + wb * Ib.w + wc * Ic.w + wd * Id.w;

        o4[col * 16 + c4] = o;
    }
}

extern "C" void kernel_launch(void* const* d_in, const int* in_sizes, int n_in,
                              void* d_out, int out_size, void* d_ws, size_t ws_size,
                              hipStream_t stream) {
    (void)in_sizes; (void)n_in; (void)out_size; (void)d_ws; (void)ws_size;
    const float* transforms = (const float*)d_in[0];   // (B,3) f32
    const float* imgs       = (const float*)d_in[1];   // (B,H,W,C) f32
    float*       out        = (float*)d_out;           // (B,256,256,C) f32

    dim3 grid(HH, BB);     // one workgroup per (batch, output row)
    dim3 block(256);       // 8 wave32 waves
    size_t shmem = 2 * (size_t)ROW_BYTES;  // 128 KB dynamic LDS (two source rows)
    hipLaunchKernelGGL(LocScaleTransform_bilinear_kernel, grid, block, shmem,
                       stream, transforms, imgs, out);
}